// GRUEncoderNetwork_74071005987300
// MI455X (gfx1250) — compile-verified
//
#include <hip/hip_runtime.h>

// GNN encoder (GRUEncoderNetwork) for MI455X / gfx1250.
// fp32 WMMA (v_wmma_f32_16x16x4_f32) for every GEMM; atomics for scatter-add.

#define N_NODES 50000
#define FDIM    128
#define EDIM    128
#define RLAYERS 4
#define GGRAPHS 512
#define NEDGES  600000

typedef float v2f __attribute__((ext_vector_type(2)));
typedef float v8f __attribute__((ext_vector_type(8)));

// LDS position (in dwords) of B element (k, col): k-pair interleaved so the
// fragment {B[k][c], B[k+1][c]} is one aligned b64 load; 16-col rotation per
// pair parity keeps the two half-waves in disjoint bank windows. 64KB total.
__device__ __forceinline__ int bpos(int k, int j) {
  int p = k >> 1;
  int cc = (j + ((p & 1) << 4)) & 127;
  return (p << 8) + (cc << 1) + (k & 1);
}

// ------------------------------------------------------------------
// C[rows x 128] slice = A[rows x 128] * B(128x128) (+bias) (+relu)
// transB: B given as [128 out][128 in] row-major -> use B^T
// ------------------------------------------------------------------
__global__ __launch_bounds__(256) void gemm_wmma_f32(
    const float* __restrict__ A, const float* __restrict__ B,
    const float* __restrict__ bias, float* __restrict__ C,
    int nrows, int ldc, int coff, int transB, int act) {
  __shared__ float Bs[128 * 128];
  const int tid = threadIdx.x;
  if (transB) {
    for (int idx = tid; idx < 128 * 128; idx += 256) {
      int j = idx >> 7, k = idx & 127;        // coalesced read B[j][k]
      Bs[bpos(k, j)] = B[j * 128 + k];        // store transposed
    }
  } else {
    for (int idx = tid; idx < 128 * 128; idx += 256) {
      int k = idx >> 7, j = idx & 127;        // coalesced read B[k][j]
      Bs[bpos(k, j)] = B[k * 128 + j];
    }
  }
  __syncthreads();

  const int wave = tid >> 5;
  const int lane = tid & 31;
  const int row0 = (blockIdx.x * 8 + wave) * 16;
  if (row0 >= nrows) return;

  const int lane15 = lane & 15;
  const int hi = lane >> 4;  // 0: rows K=kb..kb+1, 1: K=kb+2..kb+3

  // A fragment (16x4 f32): lanes 0-15 -> K={kb,kb+1}; lanes 16-31 -> K={kb+2,kb+3}
  const float* aptr = A + (size_t)(row0 + lane15) * 128 + hi * 2;

  // per-lane B fragment base pointers (kb>>1 is even, so pair parity == hi)
  const v2f* bp[8];
#pragma unroll
  for (int t = 0; t < 8; ++t) {
    int cc = (t * 16 + lane15 + (hi << 4)) & 127;
    bp[t] = (const v2f*)&Bs[(hi << 8) + (cc << 1)];
  }

  v8f acc[8] = {};

#pragma unroll 4
  for (int kb = 0; kb < 128; kb += 4) {
    v2f a = *(const v2f*)(aptr + kb);
#pragma unroll
    for (int t = 0; t < 8; ++t) {
      v2f b = bp[t][kb * 64];  // pair (kb/2 + hi), cols per bpos()
      acc[t] = __builtin_amdgcn_wmma_f32_16x16x4_f32(
          false, a, false, b, (short)0, acc[t], false, false);
    }
  }

  // C fragment: VGPR e -> row = row0 + e + (lane>=16 ? 8 : 0), col = t*16 + lane%16
  const int rbase = row0 + (hi << 3);
#pragma unroll
  for (int t = 0; t < 8; ++t) {
    const int col = t * 16 + lane15;
    const float bv = bias ? bias[col] : 0.0f;
#pragma unroll
    for (int e = 0; e < 8; ++e) {
      float v = acc[t][e] + bv;
      if (act) v = fmaxf(v, 0.0f);
      C[(size_t)(rbase + e) * ldc + coff + col] = v;
    }
  }
}

// ------------------------------------------------------------------
// helpers
// ------------------------------------------------------------------
__global__ void fill_f32(float* p, float v, int n) {
  int i = blockIdx.x * blockDim.x + threadIdx.x;
  if (i < n) p[i] = v;
}

__global__ void deg_count(const int* __restrict__ dst, float* __restrict__ deg) {
  int e = blockIdx.x * blockDim.x + threadIdx.x;
  if (e < NEDGES) atomicAdd(&deg[dst[e]], 1.0f);
}

__global__ void deg_finish(const float* __restrict__ deg, float* __restrict__ dis,
                           float* __restrict__ invdeg) {
  int i = blockIdx.x * blockDim.x + threadIdx.x;
  if (i < N_NODES) {
    float d = deg[i];            // initialized to 1.0 (self loop) + edge count
    dis[i] = rsqrtf(d);
    invdeg[i] = 1.0f / d;
  }
}

// agg = xw * invdeg[n] + conv_b   (self-loop + bias folded into scatter init)
__global__ void agg_init(const float* __restrict__ xw, const float* __restrict__ invdeg,
                         const float* __restrict__ bias, float* __restrict__ agg) {
  int i = blockIdx.x * blockDim.x + threadIdx.x;
  if (i >= N_NODES * EDIM) return;
  int n = i >> 7, c = i & 127;
  agg[i] = xw[i] * invdeg[n] + bias[c];
}

// one wave per edge: 32 lanes x float4 = 128 floats
__global__ __launch_bounds__(256) void edge_scatter(
    const int* __restrict__ src, const int* __restrict__ dst,
    const float* __restrict__ dis, const float* __restrict__ xw,
    float* __restrict__ agg) {
  int e = blockIdx.x * 8 + (threadIdx.x >> 5);
  if (e >= NEDGES) return;
  int lane = threadIdx.x & 31;
  int s = src[e];
  int d = dst[e];
  float norm = dis[s] * dis[d];
  float4 v = ((const float4*)(xw + (size_t)s * EDIM))[lane];
  float* ap = agg + (size_t)d * EDIM + lane * 4;
  atomicAdd(ap + 0, v.x * norm);
  atomicAdd(ap + 1, v.y * norm);
  atomicAdd(ap + 2, v.z * norm);
  atomicAdd(ap + 3, v.w * norm);
}

__global__ void relu_inplace(float* __restrict__ p, int n) {
  int i = blockIdx.x * blockDim.x + threadIdx.x;
  if (i < n) p[i] = fmaxf(p[i], 0.0f);
}

// GRU gates from precomputed gi, gh [N x 3E]; h updated in place
__global__ void gru_gate(const float* __restrict__ gi, const float* __restrict__ gh,
                         float* __restrict__ h) {
  int i = blockIdx.x * blockDim.x + threadIdx.x;
  if (i >= N_NODES * EDIM) return;
  int n = i >> 7, c = i & 127;
  size_t b = (size_t)n * (3 * EDIM) + c;
  float r = 1.0f / (1.0f + expf(-(gi[b] + gh[b])));
  float z = 1.0f / (1.0f + expf(-(gi[b + EDIM] + gh[b + EDIM])));
  float nn = tanhf(gi[b + 2 * EDIM] + r * gh[b + 2 * EDIM]);
  float ho = h[i];
  h[i] = (1.0f - z) * nn + z * ho;
}

__global__ void pool_count(const int* __restrict__ batch, float* __restrict__ gcnt) {
  int i = blockIdx.x * blockDim.x + threadIdx.x;
  if (i < N_NODES) atomicAdd(&gcnt[batch[i]], 1.0f);
}

__global__ void pool_sum(const int* __restrict__ batch, const float* __restrict__ h,
                         float* __restrict__ gsum) {
  int i = blockIdx.x * blockDim.x + threadIdx.x;
  if (i >= N_NODES * EDIM) return;
  int n = i >> 7, c = i & 127;
  atomicAdd(&gsum[(size_t)batch[n] * EDIM + c], h[i]);
}

__global__ void pool_finish(const float* __restrict__ gsum, const float* __restrict__ gcnt,
                            float* __restrict__ gstate) {
  int i = blockIdx.x * blockDim.x + threadIdx.x;
  if (i >= GGRAPHS * EDIM) return;
  int g = i >> 7;
  gstate[i] = gsum[i] / fmaxf(gcnt[g], 1.0f);
}

// ------------------------------------------------------------------
extern "C" void kernel_launch(void* const* d_in, const int* in_sizes, int n_in,
                              void* d_out, int out_size, void* d_ws, size_t ws_size,
                              hipStream_t stream) {
  const float* x      = (const float*)d_in[0];
  const int*   eidx   = (const int*)d_in[1];
  const int*   batch  = (const int*)d_in[2];
  const float* W_in   = (const float*)d_in[3];
  const float* b_in   = (const float*)d_in[4];
  const float* conv_W = (const float*)d_in[5];
  const float* conv_b = (const float*)d_in[6];
  const float* W_ih   = (const float*)d_in[7];
  const float* W_hh   = (const float*)d_in[8];
  const float* b_ih   = (const float*)d_in[9];
  const float* b_hh   = (const float*)d_in[10];
  const float* W_out  = (const float*)d_in[11];
  const float* b_out  = (const float*)d_in[12];
  const int* src = eidx;
  const int* dst = eidx + NEDGES;

  // carve workspace (floats), 256-float aligned chunks
  float* ws = (float*)d_ws;
  size_t o = 0;
  auto carve = [&](size_t nf) { float* p = ws + o; o += (nf + 255) & ~(size_t)255; return p; };
  float* deg    = carve(N_NODES);
  float* dis    = carve(N_NODES);
  float* invdeg = carve(N_NODES);
  float* h      = carve((size_t)N_NODES * EDIM);
  float* xw     = carve((size_t)N_NODES * EDIM);
  float* agg    = carve((size_t)N_NODES * EDIM);   // also serves as conv output x
  float* gi     = carve((size_t)N_NODES * 3 * EDIM);
  float* gh     = carve((size_t)N_NODES * 3 * EDIM);
  float* gsum   = carve((size_t)GGRAPHS * EDIM);
  float* gcnt   = carve(GGRAPHS);
  float* gstate = carve((size_t)GGRAPHS * EDIM);
  (void)ws_size; (void)in_sizes; (void)n_in; (void)out_size;

  const int NB = 256;
  const int nodeElems = N_NODES * EDIM;
  dim3 blkNode((N_NODES + NB - 1) / NB);
  dim3 blkEdge((NEDGES + NB - 1) / NB);
  dim3 blkElem((nodeElems + NB - 1) / NB);
  dim3 gemmGridN(((N_NODES / 16) + 7) / 8);   // 3125 strips -> 391 blocks
  dim3 gemmGridG(((GGRAPHS / 16) + 7) / 8);   // 32 strips -> 4 blocks
  dim3 scatGrid((NEDGES + 7) / 8);            // 8 edges (waves) per block

  // degrees (init 1.0 = self loop)
  fill_f32<<<blkNode, NB, 0, stream>>>(deg, 1.0f, N_NODES);
  deg_count<<<blkEdge, NB, 0, stream>>>(dst, deg);
  deg_finish<<<blkNode, NB, 0, stream>>>(deg, dis, invdeg);

  // h = x0 = relu(x @ W_in + b_in)
  gemm_wmma_f32<<<gemmGridN, NB, 0, stream>>>(x, W_in, b_in, h,
                                              N_NODES, EDIM, 0, 0, 1);

  for (int r = 0; r < RLAYERS; ++r) {
    // xw = h @ conv_W[r]
    gemm_wmma_f32<<<gemmGridN, NB, 0, stream>>>(h, conv_W + (size_t)r * EDIM * EDIM,
                                                nullptr, xw, N_NODES, EDIM, 0, 0, 0);
    // agg = xw*invdeg + b ; scatter-add ; relu  -> conv output x (in agg)
    agg_init<<<blkElem, NB, 0, stream>>>(xw, invdeg, conv_b + (size_t)r * EDIM, agg);
    edge_scatter<<<scatGrid, NB, 0, stream>>>(src, dst, dis, xw, agg);
    relu_inplace<<<blkElem, NB, 0, stream>>>(agg, nodeElems);

    // gi = x @ W_ih^T + b_ih ; gh = h @ W_hh^T + b_hh  (3 gate slices each)
    for (int g = 0; g < 3; ++g) {
      gemm_wmma_f32<<<gemmGridN, NB, 0, stream>>>(
          agg, W_ih + (size_t)g * EDIM * EDIM, b_ih + (size_t)g * EDIM,
          gi, N_NODES, 3 * EDIM, g * EDIM, 1, 0);
      gemm_wmma_f32<<<gemmGridN, NB, 0, stream>>>(
          h, W_hh + (size_t)g * EDIM * EDIM, b_hh + (size_t)g * EDIM,
          gh, N_NODES, 3 * EDIM, g * EDIM, 1, 0);
    }
    gru_gate<<<blkElem, NB, 0, stream>>>(gi, gh, h);  // h <- new hidden (== x)
  }

  // scatter-mean pooling
  fill_f32<<<dim3((GGRAPHS * EDIM + NB - 1) / NB), NB, 0, stream>>>(gsum, 0.0f, GGRAPHS * EDIM);
  fill_f32<<<dim3((GGRAPHS + NB - 1) / NB), NB, 0, stream>>>(gcnt, 0.0f, GGRAPHS);
  pool_count<<<blkNode, NB, 0, stream>>>(batch, gcnt);
  pool_sum<<<blkElem, NB, 0, stream>>>(batch, h, gsum);
  pool_finish<<<dim3((GGRAPHS * EDIM + NB - 1) / NB), NB, 0, stream>>>(gsum, gcnt, gstate);

  // out = gstate @ W_out + b_out   [512 x 128]
  gemm_wmma_f32<<<gemmGridG, NB, 0, stream>>>(gstate, W_out, b_out, (float*)d_out,
                                              GGRAPHS, 2 * 64, 0, 0, 0);
}